// Preprocessor_89129161326703
// MI455X (gfx1250) — compile-verified
//
#include <hip/hip_runtime.h>

typedef float v2f __attribute__((ext_vector_type(2)));
typedef float v8f __attribute__((ext_vector_type(8)));

#define WDIM 200
#define NPIX (WDIM * WDIM)      // 40000
#define NBATCH 16
#define NPTS 96                 // P*H = 8*12
#define NGROUPS (NPIX / 16)     // 2500 pixel groups of 16
#define ETA 4.0f

// ---------------------------------------------------------------------------
// ws[0..15]  = per-batch min (float bits as unsigned; valid since u >= 0)
// ws[16..31] = per-batch max
// ---------------------------------------------------------------------------
__global__ void init_minmax_kernel(unsigned* ws) {
    int t = threadIdx.x;
    if (t < 16)       ws[t] = 0x7F800000u;  // +inf
    else if (t < 32)  ws[t] = 0u;           // +0.0f
}

__global__ __launch_bounds__(256) void potential_kernel(
    const float* __restrict__ goal,     // [B,2]
    const float* __restrict__ pred,     // [B,8,12,2]
    const float* __restrict__ radius,   // [B,12]
    const float* __restrict__ comfort,  // [B,12]
    const float* __restrict__ pmx,      // [W,W]
    const float* __restrict__ pmy,      // [W,W]
    float* __restrict__ out,            // [B, NPIX]  (u_total, pre-normalize)
    unsigned* __restrict__ ws)
{
    const int b    = blockIdx.x;
    const int tid  = threadIdx.x;
    const int lane = tid & 31;
    const int hi   = lane >> 4;         // 0 = lanes 0-15, 1 = lanes 16-31
    const int l16  = lane & 15;
    const int wave = tid >> 5;          // 0..7

    constexpr float DECAY[8] = {1.00f, 0.90f, 0.81f, 0.73f,
                                0.66f, 0.59f, 0.53f, 0.48f};

    // Point ordering: tile t, row m:  m<8 -> (p=m,   h=2t)
    //                                 m>=8 -> (p=m-8, h=2t+1)
    // => D VGPR v (row v for lanes 0-15, row v+8 for lanes 16-31):
    //    p == v always (decay is a compile-time constant per v),
    //    h == 2t + hi  (thresholds are per-lane registers per tile).

    // A matrices: 6 tiles. 32-bit A 16x4 layout: lanes 0-15 hold K0,K1;
    // lanes 16-31 hold K2,K3; M = lane%16. Row features [-2tx,-2ty,tx^2+ty^2,1].
    v2f A[6];
    float rsqT[6], csqT[6];
#pragma unroll
    for (int t = 0; t < 6; ++t) {
        int p = l16 & 7;
        int h = 2 * t + (l16 >> 3);
        int j = p * 12 + h;                      // point index in [0,96)
        float tx = pred[(b * NPTS + j) * 2 + 0];
        float ty = pred[(b * NPTS + j) * 2 + 1];
        float tsq = tx * tx + ty * ty;
        A[t].x = hi ? tsq  : -2.0f * tx;         // K0 / K2
        A[t].y = hi ? 1.0f : -2.0f * ty;         // K1 / K3

        // thresholds for the row this lane-half consumes: h = 2t + hi
        float r = radius [b * 12 + 2 * t + hi];
        float c = comfort[b * 12 + 2 * t + hi];
        rsqT[t] = r * r;
        csqT[t] = c * c;
    }

    const float gx = goal[b * 2 + 0];
    const float gy = goal[b * 2 + 1];

    float lmin = 3.0e38f, lmax = 0.0f;

    for (int g = blockIdx.y * 8 + wave; g < NGROUPS; g += gridDim.y * 8) {
        const int q   = g * 16 + l16;       // pixel index (both halves same pixel)
        const float px  = pmx[q];
        const float py  = pmy[q];
        const float psq = px * px + py * py;

        // B matrix 4x16: rows [px, py, 1, px^2+py^2];
        // lanes 0-15 hold K0,K1; lanes 16-31 hold K2,K3; N = lane%16.
        v2f Bm;
        Bm.x = hi ? 1.0f : px;
        Bm.y = hi ? psq  : py;

        float uo = 0.0f;
#pragma unroll
        for (int t = 0; t < 6; ++t) {
            v8f cz = {};
            // D[m][n] = (px-tx)^2 + (py-ty)^2 for point m, pixel n
            v8f d = __builtin_amdgcn_wmma_f32_16x16x4_f32(
                false, A[t], false, Bm, (short)0, cz, false, false);
            const float rs = rsqT[t];
            const float cs = csqT[t];
#pragma unroll
            for (int v = 0; v < 8; ++v) {
                const float ed  = 4.0f * DECAY[v];   // ETA * decay[p=v]
                const float hed = 2.0f * DECAY[v];   // 0.5*ETA * decay
                const float dq2 = d[v];
                // nested where == max of two independent selects (r <= c)
                uo = fmaxf(uo, dq2 <= rs ? ed  : 0.0f);
                uo = fmaxf(uo, dq2 <= cs ? hed : 0.0f);
            }
        }
        // lanes L and L^16 hold even-h / odd-h point subsets for the same pixel
        uo = fmaxf(uo, __shfl_xor(uo, 16, 32));

        const float dx = px - gx, dy = py - gy;
        const float u = 0.5f * sqrtf(dx * dx + dy * dy) + uo;

        lmin = fminf(lmin, u);
        lmax = fmaxf(lmax, u);

        if (lane < 16) out[b * NPIX + q] = u;
    }

    // wave-wide min/max reduction (duplicated halves are harmless)
#pragma unroll
    for (int m = 1; m < 32; m <<= 1) {
        lmin = fminf(lmin, __shfl_xor(lmin, m, 32));
        lmax = fmaxf(lmax, __shfl_xor(lmax, m, 32));
    }
    if (lane == 0) {
        atomicMin(&ws[b],      __float_as_uint(lmin));
        atomicMax(&ws[16 + b], __float_as_uint(lmax));
    }
}

__global__ void normalize_kernel(float* __restrict__ out,
                                 const unsigned* __restrict__ ws) {
    int i = blockIdx.x * blockDim.x + threadIdx.x;
    if (i >= NBATCH * NPIX) return;
    int b = i / NPIX;
    float mn = __uint_as_float(ws[b]);
    float mx = __uint_as_float(ws[16 + b]);
    out[i] = (out[i] - mn) / (mx - mn);
}

extern "C" void kernel_launch(void* const* d_in, const int* in_sizes, int n_in,
                              void* d_out, int out_size, void* d_ws, size_t ws_size,
                              hipStream_t stream) {
    const float* goal    = (const float*)d_in[0];
    const float* pred    = (const float*)d_in[1];
    const float* radius  = (const float*)d_in[2];
    const float* comfort = (const float*)d_in[3];
    const float* pmx     = (const float*)d_in[4];
    const float* pmy     = (const float*)d_in[5];
    float* out = (float*)d_out;
    unsigned* ws = (unsigned*)d_ws;

    init_minmax_kernel<<<1, 32, 0, stream>>>(ws);
    potential_kernel<<<dim3(NBATCH, 64), 256, 0, stream>>>(
        goal, pred, radius, comfort, pmx, pmy, out, ws);
    normalize_kernel<<<(NBATCH * NPIX + 255) / 256, 256, 0, stream>>>(out, ws);
}